// AttentionContext_11647951307333
// MI455X (gfx1250) — compile-verified
//
#include <hip/hip_runtime.h>

#define B_   32
#define T_   2048
#define SIN_ 1024
#define HIN_ 1024
#define KD_  512
#define VD_  512
#define EPS_ 1e-12f
#define TCH_ 64                    // timesteps per ctx-partial chunk
#define NTC_ (T_ / TCH_)           // 32 chunks

typedef __attribute__((ext_vector_type(2))) float v2f;
typedef __attribute__((ext_vector_type(8))) float v8f;

// ---------------------------------------------------------------------------
// K1: query[b,k] = dot(decoder_state[b,:], Ws[k,:]) + bs[k]   (tiny)
// ---------------------------------------------------------------------------
__global__ void query_kernel(const float* __restrict__ ds, const float* __restrict__ Ws,
                             const float* __restrict__ bs, float* __restrict__ query) {
  __shared__ float s_ds[SIN_];
  const int b   = blockIdx.x;
  const int tid = threadIdx.x;
  for (int i = tid; i < SIN_; i += 256) s_ds[i] = ds[(size_t)b * SIN_ + i];
  __syncthreads();
  const int k = blockIdx.y * 256 + tid;
  const float4* w4 = (const float4*)(Ws + (size_t)k * SIN_);
  const float4* d4 = (const float4*)s_ds;
  float acc = 0.f;
#pragma unroll 4
  for (int i = 0; i < SIN_ / 4; ++i) {
    float4 w = w4[i], d = d4[i];
    acc += w.x * d.x + w.y * d.y + w.z * d.z + w.w * d.w;
  }
  query[(size_t)b * KD_ + k] = acc + bs[k];
}

// ---------------------------------------------------------------------------
// WMMA f32 GEMM (NN): C[M,N] = A[M,K] * B[K,N], one wave per 16x16 C tile.
// A frag (16x4): lanes 0-15 row m0+l, K = k+0/k+1; lanes 16-31: K = k+2/k+3.
// B frag (4x16): VGPR0 rows K=k+0 (lanes 0-15) / K=k+2 (lanes 16-31), VGPR1 K=k+1/k+3.
// D (16x16): VGPR r -> row m0+r (lanes 0-15) / m0+8+r (lanes 16-31), col n0+l.
// ---------------------------------------------------------------------------
__global__ void wmma_gemm_nn(const float* __restrict__ A, const float* __restrict__ Bm,
                             float* __restrict__ C, int N, int K) {
  const int n0   = blockIdx.x * 16;
  const int m0   = blockIdx.y * 16;
  const int lane = threadIdx.x;
  const int l    = lane & 15;
  const int koff = (lane >> 4) << 1;  // 0 for lanes 0-15, 2 for lanes 16-31
  v8f acc = {};
  const float* arow = A + (size_t)(m0 + l) * K;
  for (int k = 0; k < K; k += 4) {
    const int ka = k + koff;
    v2f a, bfr;
    a.x   = arow[ka];
    a.y   = arow[ka + 1];
    bfr.x = Bm[(size_t)ka * N + n0 + l];
    bfr.y = Bm[(size_t)(ka + 1) * N + n0 + l];
    acc = __builtin_amdgcn_wmma_f32_16x16x4_f32(false, a, false, bfr, (short)0, acc,
                                                false, false);
  }
  const int mbase = m0 + ((lane >> 4) << 3);
#pragma unroll
  for (int r = 0; r < 8; ++r)
    C[(size_t)(mbase + r) * N + n0 + l] = acc[r];
}

// ---------------------------------------------------------------------------
// WMMA f32 GEMM (NT) + scaled bias: C[m,n] = sum_k A[m,k]*Bt[n,k] + bias[n]*scale[m]
// ---------------------------------------------------------------------------
__global__ void wmma_gemm_nt_bias(const float* __restrict__ A, const float* __restrict__ Bt,
                                  const float* __restrict__ bias, const float* __restrict__ scale,
                                  float* __restrict__ C, int N, int K) {
  const int n0   = blockIdx.x * 16;
  const int m0   = blockIdx.y * 16;
  const int lane = threadIdx.x;
  const int l    = lane & 15;
  const int koff = (lane >> 4) << 1;
  v8f acc = {};
  const float* arow = A + (size_t)(m0 + l) * K;
  const float* brow = Bt + (size_t)(n0 + l) * K;  // B[k][n] = Bt[n][k]
  for (int k = 0; k < K; k += 4) {
    const int ka = k + koff;
    v2f a, bfr;
    a.x   = arow[ka];
    a.y   = arow[ka + 1];
    bfr.x = brow[ka];
    bfr.y = brow[ka + 1];
    acc = __builtin_amdgcn_wmma_f32_16x16x4_f32(false, a, false, bfr, (short)0, acc,
                                                false, false);
  }
  const int   mbase = m0 + ((lane >> 4) << 3);
  const float bn    = bias[n0 + l];
#pragma unroll
  for (int r = 0; r < 8; ++r) {
    const int m = mbase + r;
    C[(size_t)m * N + n0 + l] = acc[r] + bn * scale[m];
  }
}

// ---------------------------------------------------------------------------
// K3: energy[b,t] = dot(qprime[b,:], h[b,t,:])   — streaming pass #1 (256 MB)
// block = 256 threads = 8 waves; each wave does one t (4 KB coalesced b128 reads).
// ---------------------------------------------------------------------------
__global__ void energy_kernel(const float* __restrict__ h, const float* __restrict__ qp,
                              float* __restrict__ energy) {
  __shared__ float4 s_q[HIN_ / 4];
  const int b   = blockIdx.y;
  const int tid = threadIdx.x;
  s_q[tid] = ((const float4*)(qp + (size_t)b * HIN_))[tid];
  __syncthreads();
  const int wave = tid >> 5, lane = tid & 31;
  const int t = blockIdx.x * 8 + wave;
  const float4* row = (const float4*)(h + ((size_t)b * T_ + t) * HIN_);
  float acc = 0.f;
#pragma unroll
  for (int i = 0; i < HIN_ / 4; i += 32) {
    float4 v = row[i + lane];
    float4 q = s_q[i + lane];
    acc += v.x * q.x + v.y * q.y + v.z * q.z + v.w * q.w;
  }
  for (int off = 16; off > 0; off >>= 1) acc += __shfl_down(acc, off, 32);
  if (lane == 0) energy[(size_t)b * T_ + t] = acc;
}

// ---------------------------------------------------------------------------
// K4: softmax over T + length mask (row 0 never masked) + L1 renorm.
// scale[b] = sum(attn_final) (== S/max(S,eps)), for the bv term of context.
// ---------------------------------------------------------------------------
__global__ void softmax_kernel(const float* __restrict__ energy, const int* __restrict__ len,
                               float* __restrict__ attn, float* __restrict__ scale) {
  __shared__ float red[256];
  const int b = blockIdx.x, tid = threadIdx.x;
  const float* e = energy + (size_t)b * T_;
  float v[T_ / 256];
  float vmax = -3.402823466e38f;
#pragma unroll
  for (int i = 0; i < T_ / 256; ++i) {
    v[i] = e[tid + i * 256];
    vmax = fmaxf(vmax, v[i]);
  }
  red[tid] = vmax;
  __syncthreads();
  for (int s = 128; s > 0; s >>= 1) {
    if (tid < s) red[tid] = fmaxf(red[tid], red[tid + s]);
    __syncthreads();
  }
  vmax = red[0];
  __syncthreads();

  float lsum = 0.f;
#pragma unroll
  for (int i = 0; i < T_ / 256; ++i) {
    v[i] = __expf(v[i] - vmax);
    lsum += v[i];
  }
  red[tid] = lsum;
  __syncthreads();
  for (int s = 128; s > 0; s >>= 1) {
    if (tid < s) red[tid] += red[tid + s];
    __syncthreads();
  }
  const float ssum = red[0];
  __syncthreads();

  const int L = len[b];
  float msum = 0.f;
#pragma unroll
  for (int i = 0; i < T_ / 256; ++i) {
    const int t = tid + i * 256;
    const bool keep = (b == 0) || (t < L);  // reference skips masking sample 0
    v[i] = keep ? v[i] : 0.f;
    msum += v[i];
  }
  red[tid] = msum;
  __syncthreads();
  for (int s = 128; s > 0; s >>= 1) {
    if (tid < s) red[tid] += red[tid + s];
    __syncthreads();
  }
  msum = red[0];

  const float S   = msum / ssum;                 // sum of masked softmax probs
  const float inv = 1.0f / (fmaxf(S, EPS_) * ssum);
#pragma unroll
  for (int i = 0; i < T_ / 256; ++i)
    attn[(size_t)b * T_ + tid + i * 256] = v[i] * inv;
  if (tid == 0) scale[b] = S / fmaxf(S, EPS_);
}

// ---------------------------------------------------------------------------
// K5a: partial[tc][b][:] = sum_{t in chunk tc} attn[b,t] * h[b,t,:]
// Streaming pass #2 (256 MB). Each thread owns 4 consecutive h channels
// (per-lane global_load_b128, 512 B per wave per load). Deterministic:
// no atomics, fixed-order accumulation; combined by K5b in-order sum over chunks.
// ---------------------------------------------------------------------------
__global__ void ctx_partial_kernel(const float* __restrict__ h, const float* __restrict__ attn,
                                   float* __restrict__ partial) {
  __shared__ float s_a[TCH_];
  const int tc  = blockIdx.x;
  const int b   = blockIdx.y;
  const int tid = threadIdx.x;
  const int t0  = tc * TCH_;
  if (tid < TCH_) s_a[tid] = attn[(size_t)b * T_ + t0 + tid];
  __syncthreads();

  // row = h[b][t0+t][:] viewed as 256 float4s; this thread owns float4 #tid
  const float4* base = (const float4*)(h + ((size_t)b * T_ + t0) * HIN_) + tid;
  float4 acc = make_float4(0.f, 0.f, 0.f, 0.f);
#pragma unroll 4
  for (int t = 0; t < TCH_; ++t) {
    if ((t & 7) == 0) {
      const int tp = (t + 16 < TCH_) ? t + 16 : TCH_ - 1;
      __builtin_prefetch(base + (size_t)tp * (HIN_ / 4), 0, 1);  // global_prefetch_b8
    }
    const float4 v = base[(size_t)t * (HIN_ / 4)];
    const float  a = s_a[t];
    acc.x += a * v.x;
    acc.y += a * v.y;
    acc.z += a * v.z;
    acc.w += a * v.w;
  }
  ((float4*)(partial + ((size_t)tc * B_ + b) * HIN_))[tid] = acc;
}

// K5b: ctxh[b][hc] = sum_tc partial[tc][b][hc]  (fixed order -> deterministic)
__global__ void ctx_reduce_kernel(const float* __restrict__ partial, float* __restrict__ ctxh) {
  const int i = blockIdx.x * 256 + threadIdx.x;  // over B_*HIN_
  float acc = 0.f;
#pragma unroll
  for (int tc = 0; tc < NTC_; ++tc)
    acc += partial[(size_t)tc * (B_ * HIN_) + i];
  ctxh[i] = acc;
}

// ---------------------------------------------------------------------------
extern "C" void kernel_launch(void* const* d_in, const int* in_sizes, int n_in,
                              void* d_out, int out_size, void* d_ws, size_t ws_size,
                              hipStream_t stream) {
  const float* ds  = (const float*)d_in[0];  // [32,1024]
  const float* h   = (const float*)d_in[1];  // [32,2048,1024]
  const int*   len = (const int*)d_in[2];    // [32]
  const float* Ws  = (const float*)d_in[3];  // [512,1024]
  const float* bs  = (const float*)d_in[4];  // [512]
  const float* Wh  = (const float*)d_in[5];  // [512,1024]
  // d_in[6] = bh: constant shift along T -> softmax-invariant, unused
  const float* Wv  = (const float*)d_in[7];  // [512,1024]
  const float* bv  = (const float*)d_in[8];  // [512]

  float* out     = (float*)d_out;
  float* context = out;               // [32,512]
  float* attn    = out + B_ * VD_;    // [32,2048]

  float* ws      = (float*)d_ws;
  float* query   = ws;                        // 32*512       =  16384
  float* qprime  = query + B_ * KD_;          // 32*1024      =  32768
  float* energy  = qprime + B_ * HIN_;        // 32*2048      =  65536
  float* ctxh    = energy + B_ * T_;          // 32*1024      =  32768
  float* scale   = ctxh + B_ * HIN_;          // 32
  float* partial = scale + B_;                // 32*32*1024   = 1048576 (4 MB)

  // 1) query = ds @ Ws^T + bs
  query_kernel<<<dim3(B_, KD_ / 256), 256, 0, stream>>>(ds, Ws, bs, query);
  // 2) qprime = query @ Wh   ([32,512]x[512,1024], fp32 WMMA)
  wmma_gemm_nn<<<dim3(HIN_ / 16, B_ / 16), 32, 0, stream>>>(query, Wh, qprime, HIN_, KD_);
  // 3) energy[b,t] = qprime[b] . h[b,t]   (streaming pass 1)
  energy_kernel<<<dim3(T_ / 8, B_), 256, 0, stream>>>(h, qprime, energy);
  // 4) softmax + mask + L1 renorm -> attn output, scale[b]
  softmax_kernel<<<B_, 256, 0, stream>>>(energy, len, attn, scale);
  // 5) ctxh[b] = sum_t attn[b,t] * h[b,t]  (streaming pass 2, deterministic)
  ctx_partial_kernel<<<dim3(NTC_, B_), 256, 0, stream>>>(h, attn, partial);
  ctx_reduce_kernel<<<(B_ * HIN_ + 255) / 256, 256, 0, stream>>>(partial, ctxh);
  // 6) context = ctxh @ Wv^T + bv*scale   ([32,1024]x[1024,512], fp32 WMMA)
  wmma_gemm_nt_bias<<<dim3(VD_ / 16, B_ / 16), 32, 0, stream>>>(ctxh, Wv, bv, scale,
                                                                context, VD_, HIN_);
}